// SVDHead_71614284694314
// MI455X (gfx1250) — compile-verified
//
#include <hip/hip_runtime.h>
#include <math.h>

#define B_   16
#define DK_  512
#define N_   1024
#define M_   1024
#define EPS_ 1e-8f

typedef float v2f __attribute__((ext_vector_type(2)));
typedef float v8f __attribute__((ext_vector_type(8)));

// ---------------------------------------------------------------------------
// Kernel 0: zero the u/v potential arrays
// ---------------------------------------------------------------------------
__global__ void __launch_bounds__(256) zero_uv_kernel(float* p, int n) {
    int t = blockIdx.x * blockDim.x + threadIdx.x;
    if (t < n) p[t] = 0.0f;
}

// ---------------------------------------------------------------------------
// Kernel 1: A[b,n,m] = (src_emb[b,:,n] . tgt_emb[b,:,m]) / (sqrt(DK)*temp[b])
// One wave computes a 32x64 output tile via 2x4 grid of 16x16 f32 WMMA tiles
// (A-frags reused 4x, B-frags reused 2x -> 1.5 loads per WMMA from L2).
// src_emb layout (B,DK,N): row d contiguous over n -> coalesced lane loads.
// ---------------------------------------------------------------------------
__global__ void __launch_bounds__(256) gemm_affinity_kernel(
    const float* __restrict__ S_, const float* __restrict__ T_,
    const float* __restrict__ temp, float* __restrict__ A_)
{
    const int gwave = (blockIdx.x * blockDim.x + threadIdx.x) >> 5;
    const int lane  = threadIdx.x & 31;
    const int TN = N_ / 32, TM = M_ / 64;                 // 32 x 16 tiles
    const int b   = gwave / (TN * TM);
    const int rem = gwave % (TN * TM);
    const int n0  = (rem / TM) * 32;
    const int m0  = (rem % TM) * 64;

    const float* S = S_ + (size_t)b * DK_ * N_;
    const float* T = T_ + (size_t)b * DK_ * M_;
    const int half = lane >> 4;       // lanes 16-31 hold K+2,K+3 (ISA A-layout)
    const int lm   = lane & 15;

    v8f acc[2][4] = {};
    for (int k = 0; k < DK_; k += 4) {
        const int k0 = k + 2 * half;
        v2f afrag[2], bfrag[4];
#pragma unroll
        for (int i = 0; i < 2; ++i) {
            afrag[i].x = S[(size_t)(k0    ) * N_ + n0 + 16 * i + lm];
            afrag[i].y = S[(size_t)(k0 + 1) * N_ + n0 + 16 * i + lm];
        }
#pragma unroll
        for (int j = 0; j < 4; ++j) {
            bfrag[j].x = T[(size_t)(k0    ) * M_ + m0 + 16 * j + lm];
            bfrag[j].y = T[(size_t)(k0 + 1) * M_ + m0 + 16 * j + lm];
        }
#pragma unroll
        for (int i = 0; i < 2; ++i)
#pragma unroll
            for (int j = 0; j < 4; ++j)
                acc[i][j] = __builtin_amdgcn_wmma_f32_16x16x4_f32(
                    false, afrag[i], false, bfrag[j], (short)0, acc[i][j],
                    false, false);
    }

    const float scale = 1.0f / (22.627416997969522f * temp[b]); // sqrt(512)
    float* Ab = A_ + (size_t)b * N_ * M_;
#pragma unroll
    for (int i = 0; i < 2; ++i)
#pragma unroll
        for (int j = 0; j < 4; ++j)
#pragma unroll
            for (int r = 0; r < 8; ++r)   // C/D layout: VGPR r -> rows r, r+8
                Ab[(size_t)(n0 + 16 * i + r + 8 * half) * M_ + m0 + 16 * j + lm] =
                    acc[i][j][r] * scale;
}

// ---------------------------------------------------------------------------
// Kernel 2: row potential  u_i = -logsumexp_j(A[i,j] + v_j), slack col adds
// exp(0). One wave per row; float4 lane loads (global_load_b128), online
// max-rescaled logsumexp processing 4 elements per rescale step.
// ---------------------------------------------------------------------------
__global__ void __launch_bounds__(256) row_pass_kernel(
    const float* __restrict__ A_, const float* __restrict__ v_,
    float* __restrict__ u_)
{
    const int gwave = (blockIdx.x * blockDim.x + threadIdx.x) >> 5;
    const int lane  = threadIdx.x & 31;
    const int b = gwave / N_;
    const int i = gwave % N_;
    const float* row = A_ + (size_t)b * N_ * M_ + (size_t)i * M_;
    const float* v   = v_ + b * M_;

    float m = -3.4e38f, s = 0.0f;
    for (int j = lane * 4; j < M_; j += 128) {
        const float4 av = *(const float4*)(row + j);
        const float4 vv = *(const float4*)(v + j);
        float x0 = av.x + vv.x, x1 = av.y + vv.y;
        float x2 = av.z + vv.z, x3 = av.w + vv.w;
        float m4 = fmaxf(fmaxf(x0, x1), fmaxf(x2, x3));
        float nm = fmaxf(m, m4);
        s = s * __expf(m - nm) + __expf(x0 - nm) + __expf(x1 - nm) +
            __expf(x2 - nm) + __expf(x3 - nm);
        m = nm;
    }
    for (int off = 16; off; off >>= 1) {
        float om = __shfl_xor(m, off, 32);
        float os = __shfl_xor(s, off, 32);
        float nm = fmaxf(m, om);
        s = s * __expf(m - nm) + os * __expf(om - nm);
        m = nm;
    }
    // slack column: A=0, v_M=0 -> value 0
    float nm = fmaxf(m, 0.0f);
    s = s * __expf(m - nm) + __expf(-nm);
    m = nm;
    if (lane == 0) u_[b * N_ + i] = -(m + __logf(s));
}

// ---------------------------------------------------------------------------
// Kernel 3: column potential  v_j = -logsumexp_i(A[i,j] + u_i), slack row adds
// exp(0). Thread-per-column (coalesced across j), 4 independent online
// accumulators to break the exp dependency chain.
// ---------------------------------------------------------------------------
__global__ void __launch_bounds__(256) col_pass_kernel(
    const float* __restrict__ A_, const float* __restrict__ u_,
    float* __restrict__ v_)
{
    const int t = blockIdx.x * blockDim.x + threadIdx.x;   // B_*M_ threads
    const int b = t / M_;
    const int j = t % M_;
    const float* Ab = A_ + (size_t)b * N_ * M_ + j;
    const float* u  = u_ + b * N_;

    float m[4] = {-3.4e38f, -3.4e38f, -3.4e38f, -3.4e38f};
    float s[4] = {0.0f, 0.0f, 0.0f, 0.0f};
    for (int i = 0; i < N_; i += 4) {
#pragma unroll
        for (int c = 0; c < 4; ++c) {
            float x  = Ab[(size_t)(i + c) * M_] + u[i + c];
            float nm = fmaxf(m[c], x);
            s[c] = s[c] * __expf(m[c] - nm) + __expf(x - nm);
            m[c] = nm;
        }
    }
    float M0 = fmaxf(fmaxf(m[0], m[1]), fmaxf(m[2], m[3]));
    float S0 = s[0] * __expf(m[0] - M0) + s[1] * __expf(m[1] - M0) +
               s[2] * __expf(m[2] - M0) + s[3] * __expf(m[3] - M0);
    float nm = fmaxf(M0, 0.0f);                  // slack row term
    S0 = S0 * __expf(M0 - nm) + __expf(-nm);
    v_[t] = -(nm + __logf(S0));
}

// ---------------------------------------------------------------------------
// Kernel 4: perm = exp(A+u+v); perm_norm -> d_out; fused weighted_tgt (3 rows)
// and weights. One wave per row, float4 (b128) loads/stores.
// ---------------------------------------------------------------------------
__global__ void __launch_bounds__(256) final_rows_kernel(
    const float* __restrict__ A_, const float* __restrict__ u_,
    const float* __restrict__ v_, const float* __restrict__ tgt,
    float* __restrict__ pn_out, float* __restrict__ wt_,
    float* __restrict__ w_)
{
    const int gwave = (blockIdx.x * blockDim.x + threadIdx.x) >> 5;
    const int lane  = threadIdx.x & 31;
    const int b = gwave / N_;
    const int i = gwave % N_;
    const float* row = A_ + (size_t)b * N_ * M_ + (size_t)i * M_;
    const float* v   = v_ + b * M_;
    const float  ui  = u_[b * N_ + i];

    float rs = 0.0f;
    for (int j = lane * 4; j < M_; j += 128) {
        const float4 av = *(const float4*)(row + j);
        const float4 vv = *(const float4*)(v + j);
        rs += __expf(av.x + ui + vv.x) + __expf(av.y + ui + vv.y) +
              __expf(av.z + ui + vv.z) + __expf(av.w + ui + vv.w);
    }
    for (int off = 16; off; off >>= 1) rs += __shfl_xor(rs, off, 32);

    const float wgt = rs;                  // weights[b,i] = sum_j perm
    const float inv = 1.0f / (rs + EPS_);
    const float* tg = tgt + (size_t)b * 3 * M_;
    float* po = pn_out + (size_t)b * N_ * M_ + (size_t)i * M_;

    float a0 = 0.0f, a1 = 0.0f, a2 = 0.0f;
    for (int j = lane * 4; j < M_; j += 128) {
        const float4 av = *(const float4*)(row + j);
        const float4 vv = *(const float4*)(v + j);
        float4 pn;
        pn.x = __expf(av.x + ui + vv.x) * inv;
        pn.y = __expf(av.y + ui + vv.y) * inv;
        pn.z = __expf(av.z + ui + vv.z) * inv;
        pn.w = __expf(av.w + ui + vv.w) * inv;
        *(float4*)(po + j) = pn;
        const float4 t0 = *(const float4*)(tg + j);
        const float4 t1 = *(const float4*)(tg + M_ + j);
        const float4 t2 = *(const float4*)(tg + 2 * M_ + j);
        a0 += t0.x * pn.x + t0.y * pn.y + t0.z * pn.z + t0.w * pn.w;
        a1 += t1.x * pn.x + t1.y * pn.y + t1.z * pn.z + t1.w * pn.w;
        a2 += t2.x * pn.x + t2.y * pn.y + t2.z * pn.z + t2.w * pn.w;
    }
    for (int off = 16; off; off >>= 1) {
        a0 += __shfl_xor(a0, off, 32);
        a1 += __shfl_xor(a1, off, 32);
        a2 += __shfl_xor(a2, off, 32);
    }
    if (lane == 0) {
        wt_[(size_t)b * 3 * N_ + 0 * N_ + i] = a0;
        wt_[(size_t)b * 3 * N_ + 1 * N_ + i] = a1;
        wt_[(size_t)b * 3 * N_ + 2 * N_ + i] = a2;
        w_[b * N_ + i] = wgt;
    }
}

// ---------------------------------------------------------------------------
// Kernel 5: per-batch centroids, 3x3 cross-covariance, Procrustes via
// one-sided Jacobi SVD. One block (256 threads) per batch.
// ---------------------------------------------------------------------------
__device__ __forceinline__ float block_reduce_sum(float val, float* sm) {
    int tid = threadIdx.x;
    sm[tid] = val;
    __syncthreads();
    for (int s = 128; s > 0; s >>= 1) {
        if (tid < s) sm[tid] += sm[tid + s];
        __syncthreads();
    }
    float r = sm[0];
    __syncthreads();
    return r;
}

__device__ __forceinline__ float det3(const float X[3][3]) {
    return X[0][0] * (X[1][1] * X[2][2] - X[1][2] * X[2][1]) -
           X[0][1] * (X[1][0] * X[2][2] - X[1][2] * X[2][0]) +
           X[0][2] * (X[1][0] * X[2][1] - X[1][1] * X[2][0]);
}

__global__ void __launch_bounds__(256) procrustes_kernel(
    const float* __restrict__ src, const float* __restrict__ wt_,
    const float* __restrict__ w_, float* __restrict__ Rout,
    float* __restrict__ tout)
{
    __shared__ float sm[256];
    const int b = blockIdx.x, tid = threadIdx.x;
    const float* w  = w_  + b * N_;
    const float* sp = src + (size_t)b * 3 * N_;
    const float* wp = wt_ + (size_t)b * 3 * N_;

    float acc = 0.0f;
    for (int i = tid; i < N_; i += 256) acc += w[i];
    const float W    = block_reduce_sum(acc, sm);
    const float Winv = 1.0f / (W + EPS_);

    float sc[3] = {0, 0, 0}, tc[3] = {0, 0, 0};
    for (int i = tid; i < N_; i += 256) {
        float wn = w[i] * Winv;
#pragma unroll
        for (int d = 0; d < 3; ++d) {
            sc[d] += sp[d * N_ + i] * wn;
            tc[d] += wp[d * N_ + i] * wn;
        }
    }
#pragma unroll
    for (int d = 0; d < 3; ++d) {
        sc[d] = block_reduce_sum(sc[d], sm);
        tc[d] = block_reduce_sum(tc[d], sm);
    }

    float H[3][3] = {};
    for (int i = tid; i < N_; i += 256) {
        float wn = w[i] * Winv;
        float sv[3], cv[3];
#pragma unroll
        for (int d = 0; d < 3; ++d) {
            sv[d] = sp[d * N_ + i] - sc[d];
            cv[d] = (wp[d * N_ + i] - tc[d]) * wn;
        }
#pragma unroll
        for (int p = 0; p < 3; ++p)
#pragma unroll
            for (int q = 0; q < 3; ++q) H[p][q] += sv[p] * cv[q];
    }
#pragma unroll
    for (int p = 0; p < 3; ++p)
#pragma unroll
        for (int q = 0; q < 3; ++q) H[p][q] = block_reduce_sum(H[p][q], sm);

    if (tid == 0) {
        // one-sided Jacobi: H * Vj = B (columns of B = U * S)
        float Bm[3][3], V[3][3];
        for (int p = 0; p < 3; ++p)
            for (int q = 0; q < 3; ++q) {
                Bm[p][q] = H[p][q];
                V[p][q]  = (p == q) ? 1.0f : 0.0f;
            }
        const int pairs[3][2] = {{0, 1}, {0, 2}, {1, 2}};
        for (int sweep = 0; sweep < 20; ++sweep) {
            for (int pr = 0; pr < 3; ++pr) {
                int p = pairs[pr][0], q = pairs[pr][1];
                float al = 0, be = 0, ga = 0;
                for (int r = 0; r < 3; ++r) {
                    al += Bm[r][p] * Bm[r][p];
                    be += Bm[r][q] * Bm[r][q];
                    ga += Bm[r][p] * Bm[r][q];
                }
                if (fabsf(ga) > 1e-14f * sqrtf(al * be) && ga != 0.0f) {
                    float ze = (be - al) / (2.0f * ga);
                    float tt = ((ze >= 0.0f) ? 1.0f : -1.0f) /
                               (fabsf(ze) + sqrtf(1.0f + ze * ze));
                    float c = 1.0f / sqrtf(1.0f + tt * tt), sn = c * tt;
                    for (int r = 0; r < 3; ++r) {
                        float bp = Bm[r][p], bq = Bm[r][q];
                        Bm[r][p] = c * bp - sn * bq;
                        Bm[r][q] = sn * bp + c * bq;
                        float vp = V[r][p], vq = V[r][q];
                        V[r][p] = c * vp - sn * vq;
                        V[r][q] = sn * vp + c * vq;
                    }
                }
            }
        }
        float svn[3];
        for (int k = 0; k < 3; ++k)
            svn[k] = sqrtf(Bm[0][k] * Bm[0][k] + Bm[1][k] * Bm[1][k] +
                           Bm[2][k] * Bm[2][k]);
        // sort columns by singular value, descending (match jnp.linalg.svd)
        for (int a = 0; a < 2; ++a)
            for (int c = 0; c < 2 - a; ++c)
                if (svn[c] < svn[c + 1]) {
                    float ts = svn[c]; svn[c] = svn[c + 1]; svn[c + 1] = ts;
                    for (int r = 0; r < 3; ++r) {
                        float tb = Bm[r][c]; Bm[r][c] = Bm[r][c + 1]; Bm[r][c + 1] = tb;
                        float tv = V[r][c];  V[r][c]  = V[r][c + 1];  V[r][c + 1]  = tv;
                    }
                }
        float U[3][3];
        for (int k = 0; k < 3; ++k) {
            float inv = (svn[k] > 1e-30f) ? (1.0f / svn[k]) : 0.0f;
            for (int r = 0; r < 3; ++r) U[r][k] = Bm[r][k] * inv;
        }
        if (svn[2] <= 1e-10f * fmaxf(svn[0], 1e-30f)) {
            // rank-deficient: complete U with cross product (sign absorbed by d)
            float cx = U[1][0] * U[2][1] - U[2][0] * U[1][1];
            float cy = U[2][0] * U[0][1] - U[0][0] * U[2][1];
            float cz = U[0][0] * U[1][1] - U[1][0] * U[0][1];
            float nl = sqrtf(cx * cx + cy * cy + cz * cz) + 1e-30f;
            U[0][2] = cx / nl; U[1][2] = cy / nl; U[2][2] = cz / nl;
        }
        const float d = det3(V) * det3(U);
        float R[3][3];
        for (int p = 0; p < 3; ++p)
            for (int q = 0; q < 3; ++q)
                R[p][q] = V[p][0] * U[q][0] + V[p][1] * U[q][1] +
                          d * V[p][2] * U[q][2];
        float tv[3];
        for (int p = 0; p < 3; ++p)
            tv[p] = -(R[p][0] * sc[0] + R[p][1] * sc[1] + R[p][2] * sc[2]) + tc[p];

        for (int p = 0; p < 3; ++p) {
            for (int q = 0; q < 3; ++q) Rout[b * 9 + p * 3 + q] = R[p][q];
            tout[b * 3 + p] = tv[p];
        }
    }
}

// ---------------------------------------------------------------------------
extern "C" void kernel_launch(void* const* d_in, const int* in_sizes, int n_in,
                              void* d_out, int out_size, void* d_ws, size_t ws_size,
                              hipStream_t stream) {
    const float* srcEmb = (const float*)d_in[0];   // (B,DK,N)
    const float* tgtEmb = (const float*)d_in[1];   // (B,DK,M)
    const float* src    = (const float*)d_in[2];   // (B,3,N)
    const float* tgt    = (const float*)d_in[3];   // (B,3,M)
    const float* temp   = (const float*)d_in[4];   // (B,)

    float* out   = (float*)d_out;
    float* Rout  = out;              // 16*9  = 144
    float* tout  = out + 144;        // 16*3  = 48
    float* pnout = out + 192;        // 16*1024*1024

    // workspace layout
    float* A  = (float*)d_ws;                       // B*N*M affinity (67 MB)
    float* u  = A  + (size_t)B_ * N_ * M_;          // B*N row potentials
    float* v  = u  + B_ * N_;                       // B*M col potentials
    float* wt = v  + B_ * M_;                       // B*3*N weighted_tgt
    float* w  = wt + (size_t)B_ * 3 * N_;           // B*N weights

    zero_uv_kernel<<<(B_ * (N_ + M_) + 255) / 256, 256, 0, stream>>>(
        u, B_ * (N_ + M_));

    // 8192 waves (one per 32x64 tile), 8 waves/block
    gemm_affinity_kernel<<<1024, 256, 0, stream>>>(srcEmb, tgtEmb, temp, A);

    for (int it = 0; it < 5; ++it) {
        row_pass_kernel<<<2048, 256, 0, stream>>>(A, v, u);  // 16384 rows
        col_pass_kernel<<<64, 256, 0, stream>>>(A, u, v);    // 16384 cols
    }

    final_rows_kernel<<<2048, 256, 0, stream>>>(A, u, v, tgt, pnout, wt, w);
    procrustes_kernel<<<B_, 256, 0, stream>>>(src, wt, w, Rout, tout);
}